// GCNGraphClassifier_23880018166166
// MI455X (gfx1250) — compile-verified
//
#include <hip/hip_runtime.h>

#define N_NODES_C 100000
#define N_EDGES_C 1600000
#define IN_CH_C   128
#define HID_C     64
#define OUT_C     10
#define N_GRAPHS_C 64

typedef __attribute__((ext_vector_type(2))) float v2f;
typedef __attribute__((ext_vector_type(8))) float v8f;

// ---------------- utility kernels ----------------

__global__ __launch_bounds__(256) void k_fill(float* __restrict__ p, float v, int n) {
    int i = blockIdx.x * 256 + threadIdx.x;
    if (i < n) p[i] = v;
}

__global__ __launch_bounds__(256) void k_degree(const int* __restrict__ col,
                                                float* __restrict__ deg, int E) {
    int i = blockIdx.x * 256 + threadIdx.x;
    if (i < E) atomicAdd(&deg[col[i]], 1.0f);
}

__global__ __launch_bounds__(256) void k_rsqrt(float* __restrict__ d, int n) {
    int i = blockIdx.x * 256 + threadIdx.x;
    if (i < n) d[i] = rsqrtf(d[i]);   // deg >= 1 always (self-loops)
}

// Wt[n][k] = W[k][n]  (tiny; makes B-fragment loads contiguous b64 w/ imm offsets)
__global__ __launch_bounds__(256) void k_transpose(const float* __restrict__ W,
                                                   float* __restrict__ Wt, int K, int N) {
    int i = blockIdx.x * 256 + threadIdx.x;
    if (i < K * N) {
        int k = i / N, n = i % N;
        Wt[n * K + k] = W[i];
    }
}

// ---------------- fp32 WMMA GEMM: C[M,64] = A[M,K] @ B[K,64] ----------------
// One wave32 per 16-row M tile; the wave computes all four 16-col N tiles
// (4 x v8f accumulators). B is pre-transposed (Bt[n][k]) so both fragments are
// contiguous 8-byte loads at compile-time immediate offsets.
// Frag layouts per CDNA5 ISA (16x4 f32 A, 4x16 f32 B, 16x16 f32 C).
template <int K>
__global__ __launch_bounds__(32) void k_gemm_wmma(const float* __restrict__ A,
                                                  const float* __restrict__ Bt,
                                                  float* __restrict__ C) {
    const int N = 64;
    const int tm   = blockIdx.x;
    const int lane = threadIdx.x;
    const int half = lane >> 4;      // 0 or 1
    const int l15  = lane & 15;

    const float* Ap = A  + (size_t)(tm * 16 + l15) * K + 2 * half;  // A[m][k+2h..]
    const float* Bp = Bt + (size_t)l15 * K + 2 * half;              // Bt[n][k+2h..]

    v8f acc0 = {}, acc1 = {}, acc2 = {}, acc3 = {};
#pragma unroll
    for (int k = 0; k < K; k += 4) {
        v2f a  = *(const v2f*)(Ap + k);
        v2f b0 = *(const v2f*)(Bp + k);
        v2f b1 = *(const v2f*)(Bp + 16 * K + k);
        v2f b2 = *(const v2f*)(Bp + 32 * K + k);
        v2f b3 = *(const v2f*)(Bp + 48 * K + k);
        acc0 = __builtin_amdgcn_wmma_f32_16x16x4_f32(false, a, false, b0, (short)0, acc0, false, false);
        acc1 = __builtin_amdgcn_wmma_f32_16x16x4_f32(false, a, false, b1, (short)0, acc1, false, false);
        acc2 = __builtin_amdgcn_wmma_f32_16x16x4_f32(false, a, false, b2, (short)0, acc2, false, false);
        acc3 = __builtin_amdgcn_wmma_f32_16x16x4_f32(false, a, false, b3, (short)0, acc3, false, false);
    }

    float* Cp = C + (size_t)(tm * 16 + half * 8) * N + l15;
#pragma unroll
    for (int r = 0; r < 8; ++r) {
        Cp[(size_t)r * N +  0] = acc0[r];
        Cp[(size_t)r * N + 16] = acc1[r];
        Cp[(size_t)r * N + 32] = acc2[r];
        Cp[(size_t)r * N + 48] = acc3[r];
    }
}

// ---------------- GCN aggregation ----------------

// agg[i][:] = dinv[i]^2 * xw[i][:]   (self-loop term, direct store = init)
__global__ __launch_bounds__(256) void k_selfloop(const float4* __restrict__ xw4,
                                                  const float* __restrict__ dinv,
                                                  float4* __restrict__ agg4) {
    int i = blockIdx.x * 256 + threadIdx.x;     // over N_NODES*HID/4
    int node = i >> 4;                          // 16 float4 per node
    if (node < N_NODES_C) {
        float d = dinv[node];
        float s = d * d;
        float4 v = xw4[i];
        v.x *= s; v.y *= s; v.z *= s; v.w *= s;
        agg4[i] = v;
    }
}

// one wave per edge; lane handles a channel pair: coalesced b64 gather + f32 atomics
__global__ __launch_bounds__(256) void k_edge(const int* __restrict__ rows,
                                              const int* __restrict__ cols,
                                              const float* __restrict__ dinv,
                                              const float* __restrict__ xw,
                                              float* __restrict__ agg, int E) {
    int e = blockIdx.x * 8 + (threadIdx.x >> 5);
    int lane = threadIdx.x & 31;
    if (e < E) {
        int r = rows[e], c = cols[e];
        float nrm = dinv[r] * dinv[c];
        const float2 v = *(const float2*)&xw[(size_t)r * HID_C + 2 * lane];
        atomicAdd(&agg[(size_t)c * HID_C + 2 * lane],     nrm * v.x);
        atomicAdd(&agg[(size_t)c * HID_C + 2 * lane + 1], nrm * v.y);
    }
}

__global__ __launch_bounds__(256) void k_bias_relu(float4* __restrict__ h4,
                                                   const float* __restrict__ b, int n4) {
    int i = blockIdx.x * 256 + threadIdx.x;
    if (i < n4) {
        float4 v = h4[i];
        int base = (i & 15) * 4;
        v.x = fmaxf(v.x + b[base + 0], 0.0f);
        v.y = fmaxf(v.y + b[base + 1], 0.0f);
        v.z = fmaxf(v.z + b[base + 2], 0.0f);
        v.w = fmaxf(v.w + b[base + 3], 0.0f);
        h4[i] = v;
    }
}

// ---------------- global mean pool + classifier ----------------

__global__ __launch_bounds__(256) void k_pool(const float* __restrict__ h,
                                              const int* __restrict__ batch,
                                              float* __restrict__ sums,
                                              float* __restrict__ cnt) {
    int t = blockIdx.x * 256 + threadIdx.x;     // over N_NODES*HID
    int node = t >> 6;
    int f = t & 63;
    if (node < N_NODES_C) {
        int g = batch[node];
        atomicAdd(&sums[g * HID_C + f], h[t]);
        if (f == 0) atomicAdd(&cnt[g], 1.0f);
    }
}

__global__ __launch_bounds__(640) void k_final(const float* __restrict__ sums,
                                               const float* __restrict__ cnt,
                                               const float* __restrict__ Wl,
                                               const float* __restrict__ bl,
                                               float* __restrict__ out) {
    int t = threadIdx.x;           // 640 = 64 graphs x 10 outputs = 20 waves
    int g = t / OUT_C, o = t % OUT_C;
    float c = fmaxf(cnt[g], 1.0f);
    float acc = bl[o];
#pragma unroll
    for (int f = 0; f < HID_C; ++f)
        acc += (sums[g * HID_C + f] / c) * Wl[f * OUT_C + o];
    out[g * OUT_C + o] = acc;
}

// ---------------- launcher ----------------

extern "C" void kernel_launch(void* const* d_in, const int* in_sizes, int n_in,
                              void* d_out, int out_size, void* d_ws, size_t ws_size,
                              hipStream_t stream) {
    const float* x    = (const float*)d_in[0];
    const float* W1   = (const float*)d_in[1];
    const float* b1   = (const float*)d_in[2];
    const float* W2   = (const float*)d_in[3];
    const float* b2   = (const float*)d_in[4];
    const float* Wl   = (const float*)d_in[5];
    const float* bl   = (const float*)d_in[6];
    const int* ei     = (const int*)d_in[7];     // [2, E] flat: rows then cols
    const int* batch  = (const int*)d_in[8];
    float* out        = (float*)d_out;

    const int E = in_sizes[7] / 2;
    const int* rows = ei;
    const int* cols = ei + E;

    // workspace layout (256B-aligned blocks)
    char* ws = (char*)d_ws;
    size_t off = 0;
    float* dinv = (float*)(ws + off);  off += ((size_t)N_NODES_C * 4 + 255) & ~(size_t)255;
    float* XW   = (float*)(ws + off);  off += ((size_t)N_NODES_C * HID_C * 4 + 255) & ~(size_t)255;
    float* H    = (float*)(ws + off);  off += ((size_t)N_NODES_C * HID_C * 4 + 255) & ~(size_t)255;
    float* Wt1  = (float*)(ws + off);  off += ((size_t)IN_CH_C * HID_C * 4 + 255) & ~(size_t)255;
    float* Wt2  = (float*)(ws + off);  off += ((size_t)HID_C * HID_C * 4 + 255) & ~(size_t)255;
    float* sums = (float*)(ws + off);                       // 64*64 floats
    float* cnt  = sums + N_GRAPHS_C * HID_C;                // 64 floats (contiguous)

    const int nNH    = N_NODES_C * HID_C;                   // 6.4M elems
    const int n4     = nNH / 4;                             // 1.6M float4
    const int gNH    = (nNH + 255) / 256;
    const int g4     = (n4 + 255) / 256;
    const int gNodes = (N_NODES_C + 255) / 256;
    const int gE     = (E + 255) / 256;
    const int gE8    = (E + 7) / 8;
    const int gemmGrid = N_NODES_C / 16;                    // 6250 waves

    // degrees -> dinv ; transpose weights for contiguous B fragments
    k_fill<<<gNodes, 256, 0, stream>>>(dinv, 1.0f, N_NODES_C);     // self-loop
    k_degree<<<gE, 256, 0, stream>>>(cols, dinv, E);
    k_rsqrt<<<gNodes, 256, 0, stream>>>(dinv, N_NODES_C);
    k_transpose<<<(IN_CH_C * HID_C + 255) / 256, 256, 0, stream>>>(W1, Wt1, IN_CH_C, HID_C);
    k_transpose<<<(HID_C * HID_C + 255) / 256, 256, 0, stream>>>(W2, Wt2, HID_C, HID_C);

    // layer 1: XW = x @ W1 ; H = D^-1/2 A D^-1/2 XW ; H = relu(H + b1)
    k_gemm_wmma<IN_CH_C><<<gemmGrid, 32, 0, stream>>>(x, Wt1, XW);
    k_selfloop<<<g4, 256, 0, stream>>>((const float4*)XW, dinv, (float4*)H);
    k_edge<<<gE8, 256, 0, stream>>>(rows, cols, dinv, XW, H, E);
    k_bias_relu<<<g4, 256, 0, stream>>>((float4*)H, b1, n4);

    // layer 2: XW = H @ W2 (reuse XW); agg back into H (stream-serialized, safe)
    k_gemm_wmma<HID_C><<<gemmGrid, 32, 0, stream>>>(H, Wt2, XW);
    k_selfloop<<<g4, 256, 0, stream>>>((const float4*)XW, dinv, (float4*)H);
    k_edge<<<gE8, 256, 0, stream>>>(rows, cols, dinv, XW, H, E);
    k_bias_relu<<<g4, 256, 0, stream>>>((float4*)H, b2, n4);

    // global mean pool + classifier
    k_fill<<<(N_GRAPHS_C * HID_C + N_GRAPHS_C + 255) / 256, 256, 0, stream>>>(
        sums, 0.0f, N_GRAPHS_C * HID_C + N_GRAPHS_C);
    k_pool<<<gNH, 256, 0, stream>>>(H, batch, sums, cnt);
    k_final<<<1, 640, 0, stream>>>(sums, cnt, Wl, bl, out);
}